// Contraction_33835752358353
// MI455X (gfx1250) — compile-verified
//
#include <hip/hip_runtime.h>

// Problem constants (match reference)
#define B_   1024
#define C_   128
#define E_   10
#define L_   16
#define Q_   7
#define P3_  23
#define P2_  4
#define K3_  368           // L_*P3_
#define KTOT 384           // padded K (multiple of 8)
#define KH   192           // K half (held in registers as 48 x float2)
#define M_   1792          // Q_*L_*L_
#define USTR 388           // LDS row stride (floats) for U tiles (bank-conflict pad)
#define ZSTR 388           // LDS row stride (floats) for Z
#define TOT_IT (Q_ * 2 * L_)   // 224 linearized (w, half, xx) iterations

typedef __attribute__((ext_vector_type(2))) float v2f;
typedef __attribute__((ext_vector_type(4))) float v4f;
typedef __attribute__((ext_vector_type(8))) float v8f;
typedef __attribute__((ext_vector_type(4))) int   v4i;

// LDS layout (floats): U buffers FIRST so inner-loop A-loads get small DS immediates
#define U_OFF   0
#define XS_OFF  (U_OFF + 2 * 16 * USTR)        // 12416
#define U1_OFF  (XS_OFF + C_ * 17)             // 14592
#define Z_OFF   (U1_OFF + Q_ * L_)             // 14704
#define SMEM_FLOATS (Z_OFF + C_ * ZSTR)        // 64368
#define SMEM_BYTES  (SMEM_FLOATS * 4)          // 257472 B (<= 320KB WGP LDS)

// ---- CDNA5 async global->LDS staging (guarded; falls back to register relay) ----
#if defined(__has_builtin)
#if __has_builtin(__builtin_amdgcn_global_load_async_to_lds_b128)
#define HAVE_ASYNC 1
#endif
#endif
#ifndef HAVE_ASYNC
#define HAVE_ASYNC 0
#endif

#define AS1 __attribute__((address_space(1)))
#define AS3 __attribute__((address_space(3)))
// builtin signature (probe-confirmed): (v4i AS1*, v4i AS3*, imm offset, imm cpol)
#define GV4(p) ((AS1 v4i*)(p))
#define LV4(p) ((AS3 v4i*)(p))

#if HAVE_ASYNC
#if __has_builtin(__builtin_amdgcn_s_wait_asynccnt)
#define WAIT_ASYNC() __builtin_amdgcn_s_wait_asynccnt(0)
#else
#define WAIT_ASYNC() asm volatile("s_wait_asynccnt 0" ::: "memory")
#endif
#endif

// ---------------------------------------------------------------------------
// Kernel 1: pack Umat[m][0..383] = [ U3 row m (368 contiguous) | U2 row m (4) | 0 ]
// ---------------------------------------------------------------------------
__global__ void pack_u_kernel(const float* __restrict__ U3,
                              const float* __restrict__ U2,
                              float* __restrict__ Umat) {
    int idx = blockIdx.x * 256 + threadIdx.x;
    if (idx >= M_ * KTOT) return;
    int m  = idx / KTOT;
    int kk = idx - m * KTOT;
    float v = 0.0f;
    if (kk < K3_)            v = U3[m * K3_ + kk];
    else if (kk < K3_ + P2_) v = U2[m * P2_ + (kk - K3_)];
    Umat[idx] = v;
}

// ---------------------------------------------------------------------------
// Kernel 2: fused GEMM + epilogue.  One workgroup per node b (128 columns = C).
// ---------------------------------------------------------------------------
__global__ void __launch_bounds__(256, 1)
contraction_main(const float* __restrict__ x,  const float* __restrict__ y,
                 const float* __restrict__ U1, const float* __restrict__ w3,
                 const float* __restrict__ w2, const float* __restrict__ w1,
                 const float* __restrict__ Umat, float* __restrict__ out) {
    extern __shared__ float sm[];
    float* Us  = sm + U_OFF;
    float* XS  = sm + XS_OFF;
    float* U1s = sm + U1_OFF;
    float* Zs  = sm + Z_OFF;

    const int tid = threadIdx.x;
    const int b   = blockIdx.x;

    // ---- stage x (padded stride 17) and U1 into LDS ----
    for (int idx = tid; idx < C_ * L_; idx += 256) {
        int c = idx >> 4, i = idx & 15;
        XS[c * 17 + i] = x[(b * C_ + c) * L_ + i];
    }
    if (tid < Q_ * L_) U1s[tid] = U1[tid];

    float yv[E_];                       // uniform -> scalar loads
#pragma unroll
    for (int e = 0; e < E_; e++) yv[e] = y[b * E_ + e];
    __syncthreads();

    // ---- build Z tile: one thread per column c ----
    if (tid < C_) {
        const int c = tid;
        float wk3[P3_];
#pragma unroll
        for (int k = 0; k < P3_; k++) {
            float s = 0.0f;
#pragma unroll
            for (int e = 0; e < E_; e++) s += yv[e] * w3[(e * P3_ + k) * C_ + c];
            wk3[k] = s;
        }
        float xi[L_];
#pragma unroll
        for (int i = 0; i < L_; i++) xi[i] = XS[c * 17 + i];
        float* zr = Zs + c * ZSTR;
#pragma unroll
        for (int i = 0; i < L_; i++) {
#pragma unroll
            for (int k = 0; k < P3_; k++) zr[i * P3_ + k] = xi[i] * wk3[k];
        }
#pragma unroll
        for (int k = 0; k < P2_; k++) {
            float s = 0.0f;
#pragma unroll
            for (int e = 0; e < E_; e++) s += yv[e] * w2[(e * P2_ + k) * C_ + c];
            zr[K3_ + k] = s;
        }
#pragma unroll
        for (int kk = K3_ + P2_; kk < KTOT; kk++) zr[kk] = 0.0f;
    }
    __syncthreads();

    // ---- per-lane setup ----
    const int ws   = tid >> 5;
    const int lane = tid & 31;
    const int lo   = lane & 15;
    const int hi   = lane >> 4;
    const int ccol = ws * 16 + lo;

    float wk1 = 0.0f;
#pragma unroll
    for (int e = 0; e < E_; e++) wk1 += yv[e] * w1[e * C_ + ccol];

    float xr[8];
#pragma unroll
    for (int j = 0; j < 8; j++) xr[j] = XS[ccol * 17 + hi * 8 + j];

    const int aoff = lo * USTR + 2 * hi;           // A fragment base (floats)
    const int zoff = Z_OFF + ccol * ZSTR + 2 * hi; // B fragment base (floats)

    // U-tile staging: thread -> (row = tid/16, 24-dword chunk)
    const int prow   = tid >> 4;
    const int pchunk = (tid & 15) * 24;
#if !HAVE_ASYNC
    v4f pf[6];
#endif

    // ---- prologue: stage tile 0 into buffer 0 ----
    {
        const float* g  = Umat + prow * KTOT + pchunk;
        float* dst      = Us + prow * USTR + pchunk;
#if HAVE_ASYNC
#pragma unroll
        for (int j = 0; j < 6; j++)
            __builtin_amdgcn_global_load_async_to_lds_b128(
                GV4(g + 4 * j), LV4(dst + 4 * j), 0, 0);
        WAIT_ASYNC();
#else
#pragma unroll
        for (int j = 0; j < 6; j++) pf[j] = *(const v4f*)(g + 4 * j);
#pragma unroll
        for (int j = 0; j < 6; j++) *(v4f*)(dst + 4 * j) = pf[j];
#endif
    }
    __syncthreads();

    // ---- main loop: linear it = ((w*2 + half) * 16 + xx); tile = w*16 + xx ----
#pragma unroll 1
    for (int w = 0; w < Q_; w++) {
        float oa  = 0.0f;
        float u1a = 0.0f;
#pragma unroll 1
        for (int half = 0; half < 2; half++) {
            // B fragments for this K-half: invariant over the 16 tiles below
            v2f breg[KH / 4];
            {
                const float* Zb = sm + zoff + half * KH;
#pragma unroll
                for (int s = 0; s < KH / 4; s++) breg[s] = *(const v2f*)(Zb + 4 * s);
            }
#pragma unroll 1
            for (int xx = 0; xx < L_; xx++) {
                const int it = (w * 2 + half) * L_ + xx;

                // stage tile for it+1 into the other buffer
                if (it + 1 < TOT_IT) {
                    const int nt = (((it + 1) >> 5) << 4) + ((it + 1) & 15);
                    const float* g = Umat + (nt * 16 + prow) * KTOT + pchunk;
                    float* dst = Us + ((it + 1) & 1) * (16 * USTR) + prow * USTR + pchunk;
#if HAVE_ASYNC
#pragma unroll
                    for (int j = 0; j < 6; j++)
                        __builtin_amdgcn_global_load_async_to_lds_b128(
                            GV4(g + 4 * j), LV4(dst + 4 * j), 0, 0);
#else
#pragma unroll
                    for (int j = 0; j < 6; j++) pf[j] = *(const v4f*)(g + 4 * j);
#endif
                }

                // K-half: 48 x V_WMMA_F32_16X16X4_F32 in TWO independent
                // accumulation chains (even/odd k-step) for WMMA-pipe ILP.
                v8f acc0 = {0.f, 0.f, 0.f, 0.f, 0.f, 0.f, 0.f, 0.f};
                v8f acc1 = {0.f, 0.f, 0.f, 0.f, 0.f, 0.f, 0.f, 0.f};
                const float* Ab = Us + (it & 1) * (16 * USTR) + aoff + half * KH;
#pragma unroll
                for (int s = 0; s < KH / 8; s++) {
                    v2f a0 = *(const v2f*)(Ab + 8 * s);
                    v2f a1 = *(const v2f*)(Ab + 8 * s + 4);
                    acc0 = __builtin_amdgcn_wmma_f32_16x16x4_f32(
                        false, a0, false, breg[2 * s],     (short)0, acc0, false, false);
                    acc1 = __builtin_amdgcn_wmma_f32_16x16x4_f32(
                        false, a1, false, breg[2 * s + 1], (short)0, acc1, false, false);
                }

                // epilogue: contract over v (lane rows v = 8*hi + j), then over x
                float sv = 0.0f;
#pragma unroll
                for (int j = 0; j < 8; j++) sv += (acc0[j] + acc1[j]) * xr[j];
                sv += __shfl_xor(sv, 16, 32);
                float xsx = XS[ccol * 17 + xx];
                oa += sv * xsx;
                if (half == 0) u1a += U1s[w * 16 + xx] * xsx;

#if HAVE_ASYNC
                WAIT_ASYNC();
#else
                if (it + 1 < TOT_IT) {
                    float* dst = Us + ((it + 1) & 1) * (16 * USTR) + prow * USTR + pchunk;
#pragma unroll
                    for (int j = 0; j < 6; j++) *(v4f*)(dst + 4 * j) = pf[j];
                }
#endif
                __syncthreads();
            }
        }
        if (hi == 0) out[b * (C_ * Q_) + ccol * Q_ + w] = oa + wk1 * u1a;
    }
}

// ---------------------------------------------------------------------------
extern "C" void kernel_launch(void* const* d_in, const int* in_sizes, int n_in,
                              void* d_out, int out_size, void* d_ws, size_t ws_size,
                              hipStream_t stream) {
    (void)in_sizes; (void)n_in; (void)out_size; (void)ws_size;
    const float* x  = (const float*)d_in[0];
    const float* y  = (const float*)d_in[1];
    const float* U3 = (const float*)d_in[2];
    const float* U2 = (const float*)d_in[3];
    const float* U1 = (const float*)d_in[4];
    const float* w3 = (const float*)d_in[5];
    const float* w2 = (const float*)d_in[6];
    const float* w1 = (const float*)d_in[7];
    float* out  = (float*)d_out;
    float* Umat = (float*)d_ws;   // 1792*384*4 = 2.75 MB scratch

    (void)hipFuncSetAttribute(reinterpret_cast<const void*>(contraction_main),
                              hipFuncAttributeMaxDynamicSharedMemorySize, SMEM_BYTES);

    pack_u_kernel<<<(M_ * KTOT + 255) / 256, 256, 0, stream>>>(U3, U2, Umat);
    contraction_main<<<B_, 256, SMEM_BYTES, stream>>>(x, y, U1, w3, w2, w1, Umat, out);
}